// SignalSelfCrossAttention_69638599737582
// MI455X (gfx1250) — compile-verified
//
#include <hip/hip_runtime.h>
#include <math.h>

// ---------------------------------------------------------------------------
// SignalSelfCrossAttention for MI455X (gfx1250, wave32, WMMA)
//
// b=2, L=1024, dm=256, nh=8, dk=dv=32, 4 branches (AA, BB, AB, BA),
// complex arithmetic throughout, final complex conv1d (k=3, 1024->256 ch).
//
// All dense contractions use V_WMMA_F32_16X16X4_F32 (fp32, matches reference
// precision). One wave computes one 16x16 complex output tile with four f32
// accumulators (Prr, Pri, Pir, Pii); complex combine happens at store.
// ---------------------------------------------------------------------------

typedef __attribute__((ext_vector_type(2))) float v2f;
typedef __attribute__((ext_vector_type(8))) float v8f;

#define DM   256
#define LSEQ 1024
#define NB   2
#define NH   8
#define DK   32

// ---------------------------------------------------------------------------
// Complex GEMM:  C = alpha * (A x B), A is MxK row-major (real+imag planes),
// B is KxN row-major (TRANSB=false) or NxK row-major used as B^T (TRANSB=true).
// One wave (32 threads) per 16x16 tile. grid = (N/16, M/16, Z); per-z pointer
// offsets aZ/bZ/cZ (element strides) support batched heads.
// ---------------------------------------------------------------------------
template <bool TRANSB>
__global__ void __launch_bounds__(32)
cgemm_wmma(const float* __restrict__ Ar, const float* __restrict__ Ai,
           const float* __restrict__ Br, const float* __restrict__ Bi,
           float* __restrict__ Cr, float* __restrict__ Ci,
           int lda, int ldb, int ldc,
           long long aZ, long long bZ, long long cZ,
           int K, float alpha) {
  const int z = blockIdx.z;
  Ar += (size_t)z * aZ;  Ai += (size_t)z * aZ;
  Br += (size_t)z * bZ;  Bi += (size_t)z * bZ;
  Cr += (size_t)z * cZ;  Ci += (size_t)z * cZ;

  const int lane = threadIdx.x;        // 0..31
  const int half = lane >> 4;          // 0: K=0,1  1: K=2,3 (A layout)
  const int l15  = lane & 15;
  const int m0 = blockIdx.y * 16;
  const int n0 = blockIdx.x * 16;

  v8f prr = {}; v8f pri = {}; v8f pir = {}; v8f pii = {};

  // A 16x4 fp32 layout: lane (l15) = M row; vgpr pair = consecutive K;
  // lanes 16..31 hold K+2.  => per-lane contiguous float2 at (row, k0+2*half).
  const float* pAr = Ar + (size_t)(m0 + l15) * lda + 2 * half;
  const float* pAi = Ai + (size_t)(m0 + l15) * lda + 2 * half;

  for (int k0 = 0; k0 < K; k0 += 4) {
    v2f ar, ai, br, bi;
    ar.x = pAr[k0];  ar.y = pAr[k0 + 1];
    ai.x = pAi[k0];  ai.y = pAi[k0 + 1];

    if (TRANSB) {
      // B_eff[k][n] = B[n][k]  (B stored NxK row-major)
      const float* pBr = Br + (size_t)(n0 + l15) * ldb + k0 + 2 * half;
      const float* pBi = Bi + (size_t)(n0 + l15) * ldb + k0 + 2 * half;
      br.x = pBr[0];  br.y = pBr[1];
      bi.x = pBi[0];  bi.y = pBi[1];
    } else {
      // B 4x16 layout mirrors A: lane = N col, vgpr = K pair, lanes16+ -> K+2
      size_t r0 = (size_t)(k0 + 2 * half) * ldb + n0 + l15;
      br.x = Br[r0];  br.y = Br[r0 + ldb];
      bi.x = Bi[r0];  bi.y = Bi[r0 + ldb];
    }

    // NEG bits must be 0 for f32 WMMA (only CNeg defined) -> accumulate the
    // four real products separately and combine at store.
    prr = __builtin_amdgcn_wmma_f32_16x16x4_f32(false, ar, false, br, (short)0, prr, false, false);
    pri = __builtin_amdgcn_wmma_f32_16x16x4_f32(false, ar, false, bi, (short)0, pri, false, false);
    pir = __builtin_amdgcn_wmma_f32_16x16x4_f32(false, ai, false, br, (short)0, pir, false, false);
    pii = __builtin_amdgcn_wmma_f32_16x16x4_f32(false, ai, false, bi, (short)0, pii, false, false);
  }

  // C/D layout: VGPR v, lanes 0-15 -> (M=v, N=lane); lanes 16-31 -> (M=v+8).
#pragma unroll
  for (int v = 0; v < 8; ++v) {
    size_t idx = (size_t)(m0 + v + 8 * half) * ldc + n0 + l15;
    Cr[idx] = alpha * (prr[v] - pii[v]);
    Ci[idx] = alpha * (pri[v] + pir[v]);
  }
}

// ---------------------------------------------------------------------------
// Pack (b, dm, L) real/imag inputs into (b*L, dm) row-major planes.
// ---------------------------------------------------------------------------
__global__ void pack_transpose(const float* __restrict__ Ar, const float* __restrict__ Ai,
                               const float* __restrict__ Br, const float* __restrict__ Bi,
                               float* __restrict__ XAr, float* __restrict__ XAi,
                               float* __restrict__ XBr, float* __restrict__ XBi) {
  int gid = blockIdx.x * 256 + threadIdx.x;   // 0 .. 2*256*1024-1
  int b   = gid >> 18;                        // / (256*1024)
  int rem = gid & 262143;
  int c   = rem >> 10;
  int l   = rem & 1023;
  size_t dst = ((size_t)(b * LSEQ + l)) * DM + c;
  XAr[dst] = Ar[gid];
  XAi[dst] = Ai[gid];
  XBr[dst] = Br[gid];
  XBi[dst] = Bi[gid];
}

// ---------------------------------------------------------------------------
// MagMinMaxNorm in place on S (one batch: nh*L rows of length L).
// s *= (|s| - mn) / ((mx - mn + eps) * (|s| + eps)),  eps = 1e-12.
// ---------------------------------------------------------------------------
__global__ void mag_minmax_inplace(float* __restrict__ Sr, float* __restrict__ Si) {
  const int row = blockIdx.x;                 // h*1024 + q
  const size_t base = (size_t)row * LSEQ;
  const int c = threadIdx.x;                  // 0..255

  float mags[4];
  float mn = 3.4e38f, mx = 0.0f;
#pragma unroll
  for (int i = 0; i < 4; ++i) {
    size_t k = base + c + i * 256;
    float r = Sr[k], im = Si[k];
    float m = sqrtf(r * r + im * im);
    mags[i] = m;
    mn = fminf(mn, m);
    mx = fmaxf(mx, m);
  }

  __shared__ float smn[256], smx[256];
  smn[c] = mn; smx[c] = mx;
  __syncthreads();
  for (int s = 128; s > 0; s >>= 1) {
    if (c < s) {
      smn[c] = fminf(smn[c], smn[c + s]);
      smx[c] = fmaxf(smx[c], smx[c + s]);
    }
    __syncthreads();
  }
  mn = smn[0]; mx = smx[0];

  const float invd = 1.0f / (mx - mn + 1e-12f);
#pragma unroll
  for (int i = 0; i < 4; ++i) {
    size_t k = base + c + i * 256;
    float f = (mags[i] - mn) * invd / (mags[i] + 1e-12f);
    Sr[k] *= f;
    Si[k] *= f;
  }
}

// ---------------------------------------------------------------------------
// x = F + q_in (residual), complex whitening LayerNorm over dm=256 features,
// then store transposed into the concat/conv layout Y[b][j*256+c][l].
// ---------------------------------------------------------------------------
__global__ void residual_cln_store(const float* __restrict__ Fr, const float* __restrict__ Fi,
                                   const float* __restrict__ Xr, const float* __restrict__ Xi,
                                   const float* __restrict__ grr, const float* __restrict__ gri,
                                   const float* __restrict__ gii, const float* __restrict__ br,
                                   const float* __restrict__ bi,
                                   float* __restrict__ Yr, float* __restrict__ Yi, int jslot) {
  const int row = blockIdx.x;                 // b*1024 + l
  const int c = threadIdx.x;                  // feature 0..255
  const size_t idx = (size_t)row * DM + c;

  float xr = Fr[idx] + Xr[idx];
  float xi = Fi[idx] + Xi[idx];

  __shared__ float red[256];
  auto rsum = [&](float v) -> float {
    red[c] = v;
    __syncthreads();
    for (int s = 128; s > 0; s >>= 1) {
      if (c < s) red[c] += red[c + s];
      __syncthreads();
    }
    float r = red[0];
    __syncthreads();
    return r;
  };

  const float invn = 1.0f / (float)DM;
  float mr = rsum(xr) * invn;
  float mi = rsum(xi) * invn;
  float cr = xr - mr;
  float cm = xi - mi;
  float vrr = rsum(cr * cr) * invn + 1e-6f;
  float vii = rsum(cm * cm) * invn + 1e-6f;
  float vri = rsum(cr * cm) * invn;

  float s   = sqrtf(vrr * vii - vri * vri);
  float t   = sqrtf(vrr + vii + 2.0f * s);
  float ist = 1.0f / (s * t);
  float wrr = (vii + s) * ist;
  float wii = (vrr + s) * ist;
  float wri = -vri * ist;

  float yr = wrr * cr + wri * cm;
  float yi = wri * cr + wii * cm;
  float outr = grr[c] * yr + gri[c] * yi + br[c];
  float outi = gri[c] * yr + gii[c] * yi + bi[c];

  int b = row >> 10, l = row & 1023;
  size_t od = ((size_t)b * (4 * DM) + jslot * DM + c) * LSEQ + l;
  Yr[od] = outr;
  Yi[od] = outi;
}

// ---------------------------------------------------------------------------
// Complex conv1d: out[b][co][l] over 1024 input channels, k=3, 'same' pad.
// out_r = conv(xr,wr) - conv(xi,wi);  out_i = conv(xr,wi) + conv(xi,wr).
// Output interleaved complex64 (b, dm, L).
// ---------------------------------------------------------------------------
__global__ void cconv1d_out(const float* __restrict__ Yr, const float* __restrict__ Yi,
                            const float* __restrict__ wr, const float* __restrict__ wi,
                            float* __restrict__ out) {
  const int l  = blockIdx.x * 256 + threadIdx.x;  // 0..1023
  const int co = blockIdx.y;                      // 0..255
  const int b  = blockIdx.z;                      // 0..1
  const float* xr = Yr + (size_t)b * (4 * DM) * LSEQ;
  const float* xi = Yi + (size_t)b * (4 * DM) * LSEQ;

  float arr = 0.f, ari = 0.f, air = 0.f, aii = 0.f;
  for (int ci = 0; ci < 4 * DM; ++ci) {
    const float* w0r = wr + ((size_t)co * (4 * DM) + ci) * 3;
    const float* w0i = wi + ((size_t)co * (4 * DM) + ci) * 3;
    const float* rr = xr + (size_t)ci * LSEQ;
    const float* ri = xi + (size_t)ci * LSEQ;
    float xmr = (l > 0) ? rr[l - 1] : 0.0f;
    float x0r = rr[l];
    float xpr = (l < LSEQ - 1) ? rr[l + 1] : 0.0f;
    float xmi = (l > 0) ? ri[l - 1] : 0.0f;
    float x0i = ri[l];
    float xpi = (l < LSEQ - 1) ? ri[l + 1] : 0.0f;
    float w0 = w0r[0], w1 = w0r[1], w2 = w0r[2];
    float u0 = w0i[0], u1 = w0i[1], u2 = w0i[2];
    arr = fmaf(w0, xmr, arr); arr = fmaf(w1, x0r, arr); arr = fmaf(w2, xpr, arr);
    air = fmaf(w0, xmi, air); air = fmaf(w1, x0i, air); air = fmaf(w2, xpi, air);
    ari = fmaf(u0, xmr, ari); ari = fmaf(u1, x0r, ari); ari = fmaf(u2, xpr, ari);
    aii = fmaf(u0, xmi, aii); aii = fmaf(u1, x0i, aii); aii = fmaf(u2, xpi, aii);
  }
  size_t idx = ((size_t)b * DM + co) * LSEQ + l;
  out[2 * idx]     = arr - aii;
  out[2 * idx + 1] = ari + air;
}

// ---------------------------------------------------------------------------
extern "C" void kernel_launch(void* const* d_in, const int* in_sizes, int n_in,
                              void* d_out, int out_size, void* d_ws, size_t ws_size,
                              hipStream_t stream) {
  (void)in_sizes; (void)n_in; (void)out_size; (void)ws_size;

  const float* A_real  = (const float*)d_in[0];
  const float* A_imag  = (const float*)d_in[1];
  const float* B_real  = (const float*)d_in[2];
  const float* B_imag  = (const float*)d_in[3];
  const float* Wq_r    = (const float*)d_in[4];
  const float* Wq_i    = (const float*)d_in[5];
  const float* Wk_r    = (const float*)d_in[6];
  const float* Wk_i    = (const float*)d_in[7];
  const float* Wv_r    = (const float*)d_in[8];
  const float* Wv_i    = (const float*)d_in[9];
  const float* Wfc_r   = (const float*)d_in[10];
  const float* Wfc_i   = (const float*)d_in[11];
  const float* g_rr    = (const float*)d_in[12];
  const float* g_ri    = (const float*)d_in[13];
  const float* g_ii    = (const float*)d_in[14];
  const float* b_r     = (const float*)d_in[15];
  const float* b_i     = (const float*)d_in[16];
  const float* conv_wr = (const float*)d_in[17];
  const float* conv_wi = (const float*)d_in[18];

  float* ws = (float*)d_ws;
  const size_t P = (size_t)NB * LSEQ * DM;      // 524288 elements per plane
  float* XAr = ws + 0 * P;   float* XAi = ws + 1 * P;
  float* XBr = ws + 2 * P;   float* XBi = ws + 3 * P;
  float* Qr  = ws + 4 * P;   float* Qi  = ws + 5 * P;
  float* Kr  = ws + 6 * P;   float* Ki  = ws + 7 * P;
  float* Vr  = ws + 8 * P;   float* Vi  = ws + 9 * P;
  float* Or  = ws + 10 * P;  float* Oi  = ws + 11 * P;
  float* Fr  = ws + 12 * P;  float* Fi  = ws + 13 * P;
  float* Yr  = ws + 14 * P;                     // 4 planes (b, 1024, L)
  float* Yi  = ws + 18 * P;                     // 4 planes
  float* Sr  = ws + 22 * P;                     // nh*L*L = 16 planes (one batch)
  float* Si  = ws + 38 * P;                     // 16 planes; total = 54*P floats

  // Pack inputs (b, dm, L) -> (b*L, dm)
  pack_transpose<<<2048, 256, 0, stream>>>(A_real, A_imag, B_real, B_imag,
                                           XAr, XAi, XBr, XBi);

  const dim3 gProj(DM / 16, (NB * LSEQ) / 16, 1);   // 16 x 128
  const dim3 gScore(LSEQ / 16, LSEQ / 16, NH);      // 64 x 64 x 8
  const dim3 gAV(DK / 16, LSEQ / 16, NH);           // 2 x 64 x 8
  const float scale = 1.0f / sqrtf((float)DK);

  for (int j = 0; j < 4; ++j) {
    // branch order: 0=AA, 1=BB, 2=AB (q=A,kv=B), 3=BA (q=B,kv=A)
    const float* Xq_r  = (j == 0 || j == 2) ? XAr : XBr;
    const float* Xq_i  = (j == 0 || j == 2) ? XAi : XBi;
    const float* Xkv_r = (j == 0 || j == 3) ? XAr : XBr;
    const float* Xkv_i = (j == 0 || j == 3) ? XAi : XBi;

    const size_t wOff = (size_t)j * DM * DM;

    // Q/K/V projections (2048x256 @ 256x256 complex)
    cgemm_wmma<false><<<gProj, 32, 0, stream>>>(Xq_r, Xq_i, Wq_r + wOff, Wq_i + wOff,
                                                Qr, Qi, DM, DM, DM, 0, 0, 0, DM, 1.0f);
    cgemm_wmma<false><<<gProj, 32, 0, stream>>>(Xkv_r, Xkv_i, Wk_r + wOff, Wk_i + wOff,
                                                Kr, Ki, DM, DM, DM, 0, 0, 0, DM, 1.0f);
    cgemm_wmma<false><<<gProj, 32, 0, stream>>>(Xkv_r, Xkv_i, Wv_r + wOff, Wv_i + wOff,
                                                Vr, Vi, DM, DM, DM, 0, 0, 0, DM, 1.0f);

    // Attention per batch (keeps 64 MB score working set L2-resident)
    for (int b = 0; b < NB; ++b) {
      const size_t qoff = (size_t)b * LSEQ * DM;
      // S[h] = (Q_h @ K_h^T) / sqrt(dk): z=head, col offset 32 per head
      cgemm_wmma<true><<<gScore, 32, 0, stream>>>(
          Qr + qoff, Qi + qoff, Kr + qoff, Ki + qoff, Sr, Si,
          DM, DM, LSEQ, DK, DK, (long long)LSEQ * LSEQ, DK, scale);
      mag_minmax_inplace<<<NH * LSEQ, 256, 0, stream>>>(Sr, Si);
      // O_h = S[h] @ V_h
      cgemm_wmma<false><<<gAV, 32, 0, stream>>>(
          Sr, Si, Vr + qoff, Vi + qoff, Or + qoff, Oi + qoff,
          LSEQ, DM, DM, (long long)LSEQ * LSEQ, DK, DK, LSEQ, 1.0f);
    }

    // FC projection
    cgemm_wmma<false><<<gProj, 32, 0, stream>>>(Or, Oi, Wfc_r + wOff, Wfc_i + wOff,
                                                Fr, Fi, DM, DM, DM, 0, 0, 0, DM, 1.0f);

    // Residual + complex LayerNorm, store transposed into concat slot j
    residual_cln_store<<<NB * LSEQ, 256, 0, stream>>>(
        Fr, Fi, Xq_r, Xq_i,
        g_rr + j * DM, g_ri + j * DM, g_ii + j * DM, b_r + j * DM, b_i + j * DM,
        Yr, Yi, j);
  }

  // Final complex conv1d -> interleaved complex64 (b, dm, L)
  const dim3 gConv(LSEQ / 256, DM, NB);
  cconv1d_out<<<gConv, 256, 0, stream>>>(Yr, Yi, conv_wr, conv_wi, (float*)d_out);
}